// SOPL_Layer_34196529611296
// MI455X (gfx1250) — compile-verified
//
#include <hip/hip_runtime.h>
#include <math.h>

// MI455X / gfx1250, wave32. Fused soft-kmeans with beta time prior.
// f32 WMMA 16x16x4 for both GEMMs; x (102MB) is L2-resident (192MB L2).
// x tiles staged to LDS via Tensor Data Mover (tensor_load_to_lds) when the
// builtin is available; weight [N,K] never materialized to memory.

typedef float v2f __attribute__((ext_vector_type(2)));
typedef float v8f __attribute__((ext_vector_type(8)));
typedef unsigned int u32x4 __attribute__((ext_vector_type(4)));
typedef int i32x4 __attribute__((ext_vector_type(4)));
typedef int i32x8 __attribute__((ext_vector_type(8)));

#define KK 64
#define DD 64
#define ROWP 68              // padded LDS row stride (floats): 16B-aligned rows,
                             // 8B-aligned (even) fragment pairs, bank-skewed
#define EPSV 1e-10f
#define PMAX 69.0775527898f  // -log(1e-30)

#if defined(__has_builtin)
#if __has_builtin(__builtin_amdgcn_tensor_load_to_lds)
#define SK_HAVE_TDM 1
#endif
#endif
#ifndef SK_HAVE_TDM
#define SK_HAVE_TDM 0
#endif

// ---------------- init: centers <- centers0, time stats <- 0 ----------------
__global__ void sk_init(const float* __restrict__ c0, float* __restrict__ cent,
                        float* __restrict__ tm, float* __restrict__ tv) {
  int i = blockIdx.x * blockDim.x + threadIdx.x;
  if (i < KK * DD) cent[i] = c0[i];
  if (i < KK) { tm[i] = 0.f; tv[i] = 0.f; }
}

// ---------------- xsq[n] = sum_d x[n][d]^2 ----------------
__global__ void sk_xsq(const float* __restrict__ x, float* __restrict__ xsq, int N) {
  int n = blockIdx.x * blockDim.x + threadIdx.x;
  if (n < N) {
    const float4* xr = (const float4*)(x + (size_t)n * DD);
    float s = 0.f;
#pragma unroll
    for (int i = 0; i < DD / 4; ++i) {
      float4 v = xr[i];
      s += v.x * v.x + v.y * v.y + v.z * v.z + v.w * v.w;
    }
    xsq[n] = s;
  }
}

// ---------------- per-cluster beta params ----------------
// karr: [0:64)=a-1, [64:128)=b-1, [128:192)=Cc, [192:256)=csq
// kaux: [0:64)=betaln, [64:128)=ok flag
__global__ void sk_kprep(const float* __restrict__ tm, const float* __restrict__ tv,
                         float* __restrict__ karr, float* __restrict__ kaux,
                         float* __restrict__ S) {
  int k = threadIdx.x;
  if (k < KK) {
    float mu = tm[k], var = tv[k];
    float sigma = sqrtf(var);
    bool valid = (sigma > 1e-4f) && (mu > 0.f);
    float mu_s = valid ? mu : 0.5f;
    float var_s = valid ? var : 1.f;
    float a = (1.f - mu_s) * mu_s * mu_s / var_s - mu_s;
    float b = a / mu_s - a;
    bool ok = valid && (a > 0.f) && (b > 0.f);
    a = ok ? a : 1.f;
    b = ok ? b : 1.f;
    float LB = lgammaf(a) + lgammaf(b) - lgammaf(a + b);
    karr[k] = a - 1.f;
    karr[KK + k] = b - 1.f;
    kaux[k] = LB;
    kaux[KK + k] = ok ? 1.f : 0.f;
    S[k] = 0.f;
  }
}

// ---------------- S_k = sum_n exp((a-1)log t + (b-1)log1p(-t) - betaln) ----------------
__global__ __launch_bounds__(256) void sk_sumS(const float* __restrict__ karr,
                                               const float* __restrict__ kaux,
                                               float* __restrict__ S, int N) {
  __shared__ float sS[KK];
  int tid = threadIdx.x;
  for (int i = tid; i < KK; i += 256) sS[i] = 0.f;
  __syncthreads();
  int k = tid & 63;
  int sub = tid >> 6;  // 0..3
  float A1 = karr[k], B1 = karr[KK + k], LB = kaux[k], okf = kaux[KK + k];
  float local = 0.f;
  if (okf > 0.5f) {
    float invN = 1.f / (float)N;
    for (int n = blockIdx.x * 4 + sub; n < N; n += gridDim.x * 4) {
      float t = ((float)n + 0.5f) * invN;
      float lp = A1 * __logf(t) + B1 * log1pf(-t) - LB;
      local += __expf(lp);
    }
  }
  atomicAdd(&sS[k], local);
  __syncthreads();
  if (tid < KK) atomicAdd(&S[tid], sS[tid]);
}

// ---------------- Cc_k, csq_k; zero accumulators ----------------
__global__ __launch_bounds__(256) void sk_kprep2(const float* __restrict__ kaux,
                                                 const float* __restrict__ S,
                                                 const float* __restrict__ cent,
                                                 float* __restrict__ karr,
                                                 float* __restrict__ cent_acc,
                                                 float* __restrict__ stats, int N) {
  int tid = threadIdx.x;
  for (int i = tid; i < KK * DD; i += 256) cent_acc[i] = 0.f;
  for (int i = tid; i < 3 * KK; i += 256) stats[i] = 0.f;
  if (tid < KK) {
    float ok = kaux[KK + tid];
    float Cc;
    if (ok > 0.5f) {
      Cc = kaux[tid] + __logf(fmaxf(S[tid], EPSV));  // betaln + log(sum)
    } else {
      Cc = __logf((float)N);  // uniform 1/N prior
      karr[tid] = 0.f;        // A1 = 0
      karr[KK + tid] = 0.f;   // B1 = 0 -> prior == clamp(logN,0,PMAX) == logN
    }
    karr[2 * KK + tid] = Cc;
    float cs = 0.f;
    const float* cr = cent + (size_t)tid * DD;
#pragma unroll 8
    for (int d = 0; d < DD; ++d) { float c = cr[d]; cs += c * c; }
    karr[3 * KK + tid] = cs;
  }
}

// ---- TDM: DMA one 64x64 f32 tile (row stride 64) into LDS with +4dw row pad ----
__device__ __forceinline__ void sk_tdm_load_tile(const float* gsrc, unsigned ldsOff) {
#if SK_HAVE_TDM
  unsigned long long ga = (unsigned long long)(const void*)gsrc;
  u32x4 g0;
  g0.x = 1u;                                          // count=1, user descriptor
  g0.y = ldsOff;                                      // lds_addr (bytes)
  g0.z = (unsigned)(ga & 0xffffffffull);              // global_addr[31:0]
  g0.w = (unsigned)((ga >> 32) & 0x01ffffffull)       // global_addr[56:32]
         | (2u << 30);                                // type = 2 ("image")
  // data_size=4B(2), pad_enable, pad_interval=64dw(5), pad_amount=4dw(3)
  i32x8 g1;
  g1[0] = (int)((2u << 16) | (1u << 20) | (5u << 22) | (3u << 25));
  g1[1] = (int)(64u << 16);  // tensor_dim0[15:0]=64 (bits 79:48 low half)
  g1[2] = (int)(64u << 16);  // tensor_dim0 hi=0; tensor_dim1[15:0]=64
  g1[3] = (int)(64u << 16);  // tensor_dim1 hi=0; tile_dim0=64
  g1[4] = 64;                // tile_dim1=64; tile_dim2=0
  g1[5] = 64;                // tensor_dim0_stride[31:0]=64
  g1[6] = 0;                 // stride0 hi=0; tensor_dim1_stride lo=0
  g1[7] = 0;
  i32x4 z4 = {0, 0, 0, 0};   // 2-D tile: groups 2/3 unused
#if __clang_major__ >= 23
  i32x8 z8 = {0, 0, 0, 0, 0, 0, 0, 0};
  __builtin_amdgcn_tensor_load_to_lds(g0, g1, z4, z4, z8, 0);
#else
  __builtin_amdgcn_tensor_load_to_lds(g0, g1, z4, z4, 0);
#endif
#else
  (void)gsrc; (void)ldsOff;
#endif
}

// ---------------- fused main pass ----------------
// block = 128 threads = 4 waves, grid-stride over 64-row blocks.
// TDM stages the 64x64 x-tile into LDS (padded rows); GEMM1 (dist) and GEMM2
// (weight^T @ x) both run on v_wmma_f32_16x16x4_f32 out of LDS.
__global__ __launch_bounds__(128) void sk_main(
    const float* __restrict__ x, const float* __restrict__ xsq,
    const float* __restrict__ cent, const float* __restrict__ karr,
    float* __restrict__ cent_acc, float* __restrict__ stats, int N) {
  __shared__ float sC[KK * ROWP];      // centers [k][d], padded rows
  __shared__ float sX[64 * ROWP];      // x tile [n_local][d], padded rows
  __shared__ float sW[64 * (KK + 1)];  // weight tile [n_local][k]
  __shared__ float sStat[3 * KK];

  const int tid = threadIdx.x;
  const int wave = tid >> 5;
  const int lane = tid & 31;
  const int l16 = lane & 15;
  const int half = lane >> 4;
  const int koff = half * 2;
  const unsigned sXoff = (unsigned)(unsigned long long)(void*)sX;

  for (int i = tid; i < KK * 16; i += 128) {  // centers -> LDS, float4-wide
    int r = i >> 4, c4 = (i & 15) * 4;
    *(float4*)(sC + r * ROWP + c4) = *(const float4*)(cent + r * DD + c4);
  }
  for (int i = tid; i < 3 * KK; i += 128) sStat[i] = 0.f;
  __syncthreads();

  // loop-invariant per-lane cluster params: tile ct covers k = 16ct + l16
  float A1[4], B1[4], Cc[4], Cs[4];
#pragma unroll
  for (int ct = 0; ct < 4; ++ct) {
    int k = ct * 16 + l16;
    A1[ct] = karr[k];
    B1[ct] = karr[KK + k];
    Cc[ct] = karr[2 * KK + k];
    Cs[ct] = karr[3 * KK + k];
  }

  const v8f zero8 = {0.f, 0.f, 0.f, 0.f, 0.f, 0.f, 0.f, 0.f};
  v8f c2[4];  // GEMM2 accumulators: k-slice [16*wave..+16) x d-tile dt
#pragma unroll
  for (int dt = 0; dt < 4; ++dt) c2[dt] = zero8;

  const float invN = 1.f / (float)N;
  const int nBlocks = N >> 6;  // N % 64 == 0 here (400000)
  for (int blk = blockIdx.x; blk < nBlocks; blk += gridDim.x) {
    const int rowBase = blk * 64 + wave * 16;
    const float* xblk = x + (size_t)blk * 64 * DD;

#if SK_HAVE_TDM
    if (wave == 0) sk_tdm_load_tile(xblk, sXoff);  // TENSORcnt++
#endif

    // per-row scalars (independent of sX -> overlaps the DMA)
    float l1v[8], l2v[8], tval[8], xs[8];
#pragma unroll
    for (int j = 0; j < 8; ++j) {
      int n = rowBase + j + half * 8;
      float t = ((float)n + 0.5f) * invN;
      tval[j] = t;
      l1v[j] = __logf(t);
      l2v[j] = log1pf(-t);
      xs[j] = xsq[n];
    }

#if SK_HAVE_TDM
    if (wave == 0) __builtin_amdgcn_s_wait_tensorcnt(0);
#else
    for (int i = tid; i < 64 * 16; i += 128) {  // fallback: float4 staging
      int r = i >> 4, c4 = (i & 15) * 4;
      *(float4*)(sX + r * ROWP + c4) = *(const float4*)(xblk + r * DD + c4);
    }
#endif
    __syncthreads();  // x tile visible to all waves

    // ---- GEMM1: dot[n][k] (A = x rows from sX, B = centers^T from sC) ----
    v8f acc[4];
#pragma unroll
    for (int ct = 0; ct < 4; ++ct) acc[ct] = zero8;
    const float* xrow = sX + (wave * 16 + l16) * ROWP;
#pragma unroll
    for (int db = 0; db < DD; db += 4) {
      v2f a = *(const v2f*)(xrow + db + koff);
#pragma unroll
      for (int ct = 0; ct < 4; ++ct) {
        v2f b = *(const v2f*)(sC + (ct * 16 + l16) * ROWP + db + koff);
        acc[ct] = __builtin_amdgcn_wmma_f32_16x16x4_f32(
            false, a, false, b, (short)0, acc[ct], false, false);
      }
    }

    // dist = xsq + csq - 2*dot + prior ; prior = clamp(Cc - A1*l1 - B1*l2, 0, PMAX)
    float rmin[8];
#pragma unroll
    for (int j = 0; j < 8; ++j) rmin[j] = 3.4e38f;
#pragma unroll
    for (int ct = 0; ct < 4; ++ct)
#pragma unroll
      for (int j = 0; j < 8; ++j) {
        float pri = Cc[ct] - A1[ct] * l1v[j] - B1[ct] * l2v[j];
        pri = fminf(fmaxf(pri, 0.f), PMAX);
        float dd = xs[j] + Cs[ct] - 2.f * acc[ct][j] + pri;
        acc[ct][j] = dd;
        rmin[j] = fminf(rmin[j], dd);
      }
    // reductions over 64 k: 16-lane butterflies (k lives across lanes+tiles)
#pragma unroll
    for (int m = 1; m <= 8; m <<= 1)
#pragma unroll
      for (int j = 0; j < 8; ++j)
        rmin[j] = fminf(rmin[j], __shfl_xor(rmin[j], m, 32));

    float rsum[8];
#pragma unroll
    for (int j = 0; j < 8; ++j) rsum[j] = 0.f;
#pragma unroll
    for (int ct = 0; ct < 4; ++ct)
#pragma unroll
      for (int j = 0; j < 8; ++j) {
        float e = __expf(rmin[j] - acc[ct][j]);  // softmax(-dist)
        acc[ct][j] = e;
        rsum[j] += e;
      }
#pragma unroll
    for (int m = 1; m <= 8; m <<= 1)
#pragma unroll
      for (int j = 0; j < 8; ++j) rsum[j] += __shfl_xor(rsum[j], m, 32);
#pragma unroll
    for (int j = 0; j < 8; ++j) rsum[j] = 1.f / rsum[j];
#pragma unroll
    for (int ct = 0; ct < 4; ++ct)
#pragma unroll
      for (int j = 0; j < 8; ++j) acc[ct][j] *= rsum[j];  // acc = weight

    // ---- stats: wsum / wt / wt2 per k ----
#pragma unroll
    for (int ct = 0; ct < 4; ++ct) {
      float s0 = 0.f, s1 = 0.f, s2 = 0.f;
#pragma unroll
      for (int j = 0; j < 8; ++j) {
        float w = acc[ct][j];
        s0 += w;
        s1 += w * tval[j];
        s2 += w * tval[j] * tval[j];
      }
      s0 += __shfl_xor(s0, 16, 32);
      s1 += __shfl_xor(s1, 16, 32);
      s2 += __shfl_xor(s2, 16, 32);
      if (half == 0) {
        int k = ct * 16 + l16;
        atomicAdd(&sStat[k], s0);
        atomicAdd(&sStat[KK + k], s1);
        atomicAdd(&sStat[2 * KK + k], s2);
      }
    }

    // ---- stage W to LDS (transpose for GEMM2 A-fragments) ----
    __syncthreads();  // previous iteration's GEMM2 reads of sW are done
#pragma unroll
    for (int ct = 0; ct < 4; ++ct)
#pragma unroll
      for (int j = 0; j < 8; ++j) {
        int ln = wave * 16 + j + half * 8;
        sW[ln * (KK + 1) + ct * 16 + l16] = acc[ct][j];
      }
    __syncthreads();

    // ---- GEMM2: cent_acc[k][d] += sum_n W[n][k] * x[n][d] (x from sX) ----
    const int kw = wave * 16;
#pragma unroll
    for (int nc = 0; nc < 16; ++nc) {
      int n0 = nc * 4 + koff;
      v2f a;  // A' = W^T: M = k rows, Kdim = n
      a.x = sW[n0 * (KK + 1) + kw + l16];
      a.y = sW[(n0 + 1) * (KK + 1) + kw + l16];
#pragma unroll
      for (int dt = 0; dt < 4; ++dt) {
        v2f b;  // B' = x: Kdim = n, N = d
        b.x = sX[n0 * ROWP + dt * 16 + l16];
        b.y = sX[(n0 + 1) * ROWP + dt * 16 + l16];
        c2[dt] = __builtin_amdgcn_wmma_f32_16x16x4_f32(
            false, a, false, b, (short)0, c2[dt], false, false);
      }
    }
    __syncthreads();  // sX/sW free for next iteration (and next TDM issue)
  }

  // ---- flush ----
  __syncthreads();
  for (int i = tid; i < 3 * KK; i += 128) atomicAdd(&stats[i], sStat[i]);
#pragma unroll
  for (int dt = 0; dt < 4; ++dt)
#pragma unroll
    for (int j = 0; j < 8; ++j) {
      int k = wave * 16 + j + half * 8;
      int d = dt * 16 + l16;
      atomicAdd(&cent_acc[k * DD + d], c2[dt][j]);
    }
}

// ---------------- finalize: centers, time_means, time_vars ----------------
__global__ __launch_bounds__(256) void sk_finalize(
    const float* __restrict__ cent_acc, const float* __restrict__ stats,
    float* __restrict__ cent, float* __restrict__ tm, float* __restrict__ tv) {
  int tid = threadIdx.x;
  if (tid < KK) {
    float ws = stats[tid], wt = stats[KK + tid], wt2 = stats[2 * KK + tid];
    float denom = ws + EPSV;
    float mean = wt / denom;
    float var = (wt2 - 2.f * mean * wt + mean * mean * ws) / denom;
    tm[tid] = mean;
    tv[tid] = var;
  }
  for (int i = tid; i < KK * DD; i += 256) {
    int k = i >> 6;
    cent[i] = cent_acc[i] / (stats[k] + EPSV);
  }
}

// ---------------- stable argsort of time_means (K=64) + gather ----------------
__global__ __launch_bounds__(256) void sk_output(const float* __restrict__ cent,
                                                 const float* __restrict__ tm,
                                                 float* __restrict__ out) {
  __shared__ int ord[KK];
  int tid = threadIdx.x;
  if (tid < KK) {
    float mk = tm[tid];
    int r = 0;
    for (int j = 0; j < KK; ++j) {
      float mj = tm[j];
      if (mj < mk || (mj == mk && j < tid)) ++r;
    }
    ord[r] = tid;
  }
  __syncthreads();
  for (int i = tid; i < KK * DD; i += 256)
    out[i] = cent[ord[i >> 6] * DD + (i & 63)];
}

extern "C" void kernel_launch(void* const* d_in, const int* in_sizes, int n_in,
                              void* d_out, int out_size, void* d_ws, size_t ws_size,
                              hipStream_t stream) {
  const float* x = (const float*)d_in[0];
  const float* c0 = (const float*)d_in[1];
  const int N = in_sizes[0] / DD;

  float* ws = (float*)d_ws;
  float* xsq = ws;                     // N
  float* cent = xsq + N;               // 4096
  float* cent_acc = cent + KK * DD;    // 4096
  float* karr = cent_acc + KK * DD;    // 256 (A1,B1,Cc,csq)
  float* kaux = karr + 4 * KK;         // 128 (betaln, ok)
  float* Sarr = kaux + 2 * KK;         // 64
  float* stats = Sarr + KK;            // 192 (wsum, wt, wt2)
  float* tm = stats + 3 * KK;          // 64
  float* tv = tm + KK;                 // 64

  sk_init<<<16, 256, 0, stream>>>(c0, cent, tm, tv);
  sk_xsq<<<(N + 255) / 256, 256, 0, stream>>>(x, xsq, N);

  const int nBlk = N >> 6;
  const int G = nBlk < 1024 ? nBlk : 1024;
  for (int it = 0; it < 5; ++it) {
    sk_kprep<<<1, 64, 0, stream>>>(tm, tv, karr, kaux, Sarr);
    sk_sumS<<<512, 256, 0, stream>>>(karr, kaux, Sarr, N);
    sk_kprep2<<<1, 256, 0, stream>>>(kaux, Sarr, cent, karr, cent_acc, stats, N);
    sk_main<<<G, 128, 0, stream>>>(x, xsq, cent, karr, cent_acc, stats, N);
    sk_finalize<<<1, 256, 0, stream>>>(cent_acc, stats, cent, tm, tv);
  }
  sk_output<<<1, 256, 0, stream>>>(cent, tm, (float*)d_out);
}